// SPM_59072980189277
// MI455X (gfx1250) — compile-verified
//
#include <hip/hip_runtime.h>

// ---------------------------------------------------------------------------
// MI455X plan:
//   total bytes ~237 MB (dec_W2 dominates) -> roofline floor ~10.2 us @23.3TB/s
//   total FLOPs ~0.15 G -> compute is free; keep everything f32.
//   Kernels:
//     K0: zero f[128] in workspace (determinism; ws is poisoned)
//     K1: einsum via V_WMMA_F32_16X16X4_F32 (full-precision f32 matrix op)
//     K2..K5: small 450-row GEMVs, one wave per row
//     K6: 131072x450 GEMV streaming dec_W2 with non-temporal loads (NT: 236MB
//         > 192MB L2, keep h + small weights resident)
// ---------------------------------------------------------------------------

typedef float v2f __attribute__((ext_vector_type(2)));
typedef float v8f __attribute__((ext_vector_type(8)));

#define SW     4
#define PDIM   256
#define CDIM   128
#define DDIM   128
#define MDIM   450
#define OUTN   (4 * PDIM * CDIM)   // 131072

// ---------------------------------------------------------------- K0: zero f
__global__ void spm_zero_f(float* __restrict__ f) {
    f[threadIdx.x] = 0.0f;
}

// ------------------------------------------------- K1: einsum 'tpc,ti,pi,ci->i'
// grid = 64 blocks (4 t  x 16 p-tiles), block = 1 wave32.
// Per wave: for each of 8 i-tiles, accumulate s[p0:p0+16, i0:i0+16] =
//   xw[t, p0:p0+16, :] @ CE[:, i0:i0+16] with 32 K=4 f32 WMMAs, then apply
//   PE[p,i]*TwE[t,i] per the C/D VGPR layout and atomically reduce into f[i].
__global__ __launch_bounds__(32) void spm_einsum_wmma(
    const float* __restrict__ x,    // (8, P, C), we use frames 4..7
    const float* __restrict__ TwE,  // (4, D)
    const float* __restrict__ PE,   // (P, D)
    const float* __restrict__ CE,   // (C, D)
    float* __restrict__ f)          // (D)
{
    const int lane = threadIdx.x;          // 0..31
    const int t    = blockIdx.x >> 4;      // 0..3
    const int p0   = (blockIdx.x & 15) * 16;
    const int m    = lane & 15;            // row (A) / col (B) within tile
    const int half = lane >> 4;            // 0: K=0..1, 1: K=2..3
    const int koff = half * 2;

    // This lane's A row: xw[t, p0+m, :]  (frame 8-SW+t = 4+t)
    const float* xr = x + ((size_t)(4 + t) * PDIM + (p0 + m)) * CDIM;

    for (int it = 0; it < 8; ++it) {
        const int i0 = it * 16;
        v8f acc = {0.f, 0.f, 0.f, 0.f, 0.f, 0.f, 0.f, 0.f};

        for (int c0 = 0; c0 < CDIM; c0 += 4) {
            // A (16x4 f32): lane<16 -> {A[m,c0], A[m,c0+1]},
            //               lane>=16 -> {A[m,c0+2], A[m,c0+3]}   (8B aligned)
            v2f a = *(const v2f*)(xr + c0 + koff);
            // B (4x16 f32): lane<16 -> {B[c0,  i0+m], B[c0+1,i0+m]},
            //               lane>=16 -> {B[c0+2,i0+m], B[c0+3,i0+m]}
            v2f b;
            b.x = CE[(size_t)(c0 + koff    ) * DDIM + i0 + m];
            b.y = CE[(size_t)(c0 + koff + 1) * DDIM + i0 + m];

            acc = __builtin_amdgcn_wmma_f32_16x16x4_f32(
                /*neg_a=*/false, a, /*neg_b=*/false, b,
                /*c_mod=*/(short)0, acc, /*reuse_a=*/false, /*reuse_b=*/false);
        }

        // C/D layout: VGPR r, lanes 0-15 -> M=r, N=lane; lanes 16-31 -> M=r+8.
        const int i = i0 + m;
        float s = 0.0f;
#pragma unroll
        for (int r = 0; r < 8; ++r) {
            const int p = p0 + half * 8 + r;
            s = fmaf(acc[r], PE[(size_t)p * DDIM + i], s);
        }
        s *= TwE[(size_t)t * DDIM + i];
        atomicAdd(&f[i], s);
    }
}

// --------------------------------------------------------- GEMV: out = W v + b
// One wave per row; coalesced float2 lane-strided loads; shfl_xor reduction.
// NT=1 uses non-temporal loads for the streaming 236MB dec_W2 pass.
template <int RELU, int NT>
__global__ __launch_bounds__(256) void spm_gemv_f32(
    const float* __restrict__ W, const float* __restrict__ v,
    const float* __restrict__ bias, float* __restrict__ out,
    int rows, int cols)
{
    const int gw   = (int)((blockIdx.x * blockDim.x + threadIdx.x) >> 5);
    const int lane = threadIdx.x & 31;
    if (gw >= rows) return;   // wave-uniform

    const v2f* wr = (const v2f*)(W + (size_t)gw * cols);  // rows 8B aligned
    const v2f* vr = (const v2f*)v;
    const int  c2 = cols >> 1;                            // all cols are even

    float s = 0.0f;
    for (int j = lane; j < c2; j += 32) {
        v2f a = NT ? __builtin_nontemporal_load(&wr[j]) : wr[j];
        v2f xv = vr[j];
        s = fmaf(a.x, xv.x, s);
        s = fmaf(a.y, xv.y, s);
    }
#pragma unroll
    for (int off = 16; off; off >>= 1) s += __shfl_xor(s, off, 32);

    if (lane == 0) {
        float r = s + (bias ? bias[gw] : 0.0f);
        out[gw] = RELU ? fmaxf(r, 0.0f) : r;
    }
}

// ---------------------------------------------------------------------------
extern "C" void kernel_launch(void* const* d_in, const int* in_sizes, int n_in,
                              void* d_out, int out_size, void* d_ws, size_t ws_size,
                              hipStream_t stream)
{
    (void)in_sizes; (void)n_in; (void)out_size; (void)ws_size;

    const float* x   = (const float*)d_in[0];
    // d_in[1] is t (unused: window is statically the last 4 of 8 frames)
    const float* TwE = (const float*)d_in[2];
    const float* PE  = (const float*)d_in[3];
    const float* CE  = (const float*)d_in[4];
    const float* mE  = (const float*)d_in[5];
    const float* eW1 = (const float*)d_in[6];
    const float* eb1 = (const float*)d_in[7];
    const float* eW2 = (const float*)d_in[8];
    const float* eb2 = (const float*)d_in[9];
    const float* dW1 = (const float*)d_in[10];
    const float* db1 = (const float*)d_in[11];
    const float* dW2 = (const float*)d_in[12];
    const float* db2 = (const float*)d_in[13];
    float* out = (float*)d_out;

    // Workspace layout (f32, 16-float aligned slots): ~8.5 KB total
    float* ws    = (float*)d_ws;
    float* f     = ws;          // 128
    float* feat0 = ws + 128;    // 450
    float* h1    = ws + 640;    // 450
    float* feat1 = ws + 1152;   // 450
    float* h2    = ws + 1664;   // 450

    spm_zero_f<<<1, 128, 0, stream>>>(f);
    spm_einsum_wmma<<<64, 32, 0, stream>>>(x, TwE, PE, CE, f);

    const int bSmall = (MDIM * 32 + 255) / 256;   // one wave per row
    spm_gemv_f32<0, 0><<<bSmall, 256, 0, stream>>>(mE,  f,     nullptr, feat0, MDIM, DDIM);
    spm_gemv_f32<1, 0><<<bSmall, 256, 0, stream>>>(eW1, feat0, eb1,     h1,    MDIM, MDIM);
    spm_gemv_f32<0, 0><<<bSmall, 256, 0, stream>>>(eW2, h1,    eb2,     feat1, MDIM, MDIM);
    spm_gemv_f32<1, 0><<<bSmall, 256, 0, stream>>>(dW1, feat1, db1,     h2,    MDIM, MDIM);

    const int bBig = (OUTN * 32 + 255) / 256;     // 16384 blocks, 131072 waves
    spm_gemv_f32<0, 1><<<bBig, 256, 0, stream>>>(dW2, h2, db2, out, OUTN, MDIM);
}